// LocalMultiMessagePassing_81561428951695
// MI455X (gfx1250) — compile-verified
//
#include <hip/hip_runtime.h>
#include <hip/hip_bf16.h>

#define D 128
#define TWO_D 256
#define NGRAPH 16
#define NEG_INF_ORD 0x007FFFFFu   /* order-key of -inf */

typedef __attribute__((ext_vector_type(2))) float v2f;
typedef __attribute__((ext_vector_type(2))) unsigned v2u;
typedef __attribute__((ext_vector_type(8))) float v8f;

// Monotone float <-> uint mapping so we can use native u32 max atomics.
__device__ __forceinline__ unsigned f2ord(float f) {
    unsigned u = __float_as_uint(f);
    return (u & 0x80000000u) ? ~u : (u | 0x80000000u);
}
__device__ __forceinline__ float ord2f(unsigned u) {
    return (u & 0x80000000u) ? __uint_as_float(u & 0x7fffffffu)
                             : __uint_as_float(~u);
}

__global__ __launch_bounds__(256) void fill_u32(unsigned* __restrict__ p, unsigned v, int n) {
    int i = blockIdx.x * 256 + threadIdx.x;
    if (i < n) p[i] = v;
}

// One edge per 32-thread group; each lane handles 4 features (float4).
// Test-and-test-and-set: stale reads are <= true value (keys only grow), so
// skipping is always safe and extra atomics are harmless.
__global__ __launch_bounds__(256) void edge_scatter_max(
    const float* __restrict__ H, const long long* __restrict__ ei,
    unsigned* __restrict__ aggu, int nEdges)
{
    long long t = (long long)blockIdx.x * 256 + threadIdx.x;
    int e = (int)(t >> 5);
    if (e >= nEdges) return;
    int lane = (int)(t & 31);
    int src = (int)ei[e];
    int dst = (int)ei[nEdges + e];
    const float4 v = *(const float4*)(H + (size_t)src * D + lane * 4);
    unsigned* dp = aggu + (size_t)dst * D + lane * 4;
    unsigned k0 = f2ord(v.x), k1 = f2ord(v.y), k2 = f2ord(v.z), k3 = f2ord(v.w);
    if (k0 > dp[0]) atomicMax(dp + 0, k0);
    if (k1 > dp[1]) atomicMax(dp + 1, k1);
    if (k2 > dp[2]) atomicMax(dp + 2, k2);
    if (k3 > dp[3]) atomicMax(dp + 3, k3);
}

// z = leaky_relu(concat(H, decode(aggu)) @ W + b) [+ H if use_skip]
// Block: 256 threads = 8 waves. Wave w -> rows [blk*128 + w*16, +16), all 128 cols.
// fp32 WMMA 16x16x4; K = 256 in chunks of 64 staged in LDS.
__global__ __launch_bounds__(256) void gemm_step(
    const float* __restrict__ H, const unsigned* __restrict__ aggu,
    const float* __restrict__ W, const float* __restrict__ bias,
    int use_skip, float* __restrict__ Hn, int nNodes)
{
    __shared__ float sW[64 * 129];           // 64 K-rows x 128 cols, padded
    const int tid = threadIdx.x;
    const int wave = tid >> 5;
    const int lane = tid & 31;
    const int halfsel = lane >> 4;           // 0: lanes 0-15, 1: lanes 16-31
    const int lr = lane & 15;
    const int r0 = blockIdx.x * 128 + wave * 16;

    v8f acc[8];
    const v8f vzero = {0.f, 0.f, 0.f, 0.f, 0.f, 0.f, 0.f, 0.f};
#pragma unroll
    for (int i = 0; i < 8; ++i) acc[i] = vzero;

    const int rowA = min(r0 + lr, nNodes - 1);     // clamp tail rows (stores guarded)
    const float*    hrow = H    + (size_t)rowA * D;
    const unsigned* arow = aggu + (size_t)rowA * D;

    for (int kc = 0; kc < TWO_D; kc += 64) {
        __syncthreads();
        // stage W[kc..kc+63][:] -> LDS (coalesced global reads)
        for (int i = 0; i < 32; ++i) {
            int fi = tid + i * 256;                // 0..8191
            int rr = fi >> 7, cc = fi & 127;
            sW[rr * 129 + cc] = W[(size_t)(kc + rr) * D + cc];
        }
        __syncthreads();

        for (int kl = 0; kl < 64; kl += 4) {
            // A fragment: 16x4, half-waves hold K pairs {0,1} / {2,3}.
            // kk is even and rows are 512B-aligned -> single 8B load.
            int kk = kc + kl + halfsel * 2;        // wave-uniform branch below
            v2f a;
            if (kk < D) {
                a = *(const v2f*)(hrow + kk);
            } else {
                v2u u = *(const v2u*)(arow + (kk - D));
                a.x = (u.x == NEG_INF_ORD) ? 0.f : ord2f(u.x);  // empty segment -> 0
                a.y = (u.y == NEG_INF_ORD) ? 0.f : ord2f(u.y);
            }
            int brow = kl + halfsel * 2;
#pragma unroll
            for (int ct = 0; ct < 8; ++ct) {
                v2f b;
                b.x = sW[(brow    ) * 129 + ct * 16 + lr];
                b.y = sW[(brow + 1) * 129 + ct * 16 + lr];
                acc[ct] = __builtin_amdgcn_wmma_f32_16x16x4_f32(
                    false, a, false, b, (short)0, acc[ct], false, false);
            }
        }
    }

    // Epilogue: bias + leaky_relu + optional skip. C/D layout: VGPR i -> row
    // (i + 8*halfsel), col = lane&15 within the 16-col tile.
#pragma unroll
    for (int ct = 0; ct < 8; ++ct) {
        int col = ct * 16 + lr;
        float bb = bias[col];
#pragma unroll
        for (int i = 0; i < 8; ++i) {
            int row = r0 + i + halfsel * 8;
            if (row < nNodes) {
                float v = acc[ct][i] + bb;
                v = (v > 0.f) ? v : 0.01f * v;
                if (use_skip) v += H[(size_t)row * D + col];
                Hn[(size_t)row * D + col] = v;
            }
        }
    }
}

// Segment-max pool: batch_ind is sorted, so keep a running max per graph run
// and emit ~1 atomic per (block, col).
__global__ __launch_bounds__(128) void pool_max(
    const float* __restrict__ H, const long long* __restrict__ batch,
    unsigned* __restrict__ xgu, int nNodes)
{
    int col = threadIdx.x;
    int r0 = blockIdx.x * 64;
    int rend = min(r0 + 64, nNodes);
    float m = -__builtin_inff();
    int g = -1;
    for (int r = r0; r < rend; ++r) {
        int gr = (int)batch[r];
        if (gr != g) {
            if (g >= 0) atomicMax(&xgu[g * D + col], f2ord(m));
            g = gr;
            m = -__builtin_inff();
        }
        m = fmaxf(m, H[(size_t)r * D + col]);
    }
    if (g >= 0) atomicMax(&xgu[g * D + col], f2ord(m));
}

// xg = leaky_relu([xg | 0] @ wg + bg): only the first 128 rows of wg matter.
// 16x128x128 = 0.26 MFLOP -> plain VALU is fine.
__global__ __launch_bounds__(256) void xg_final(
    const unsigned* __restrict__ xgu, const float* __restrict__ wg,
    const float* __restrict__ bg, float* __restrict__ out)
{
    __shared__ float sx[NGRAPH * D];
    for (int i = threadIdx.x; i < NGRAPH * D; i += 256) {
        unsigned u = xgu[i];
        sx[i] = (u == NEG_INF_ORD) ? 0.f : ord2f(u);
    }
    __syncthreads();
    for (int o = threadIdx.x; o < NGRAPH * D; o += 256) {
        int g = o >> 7, c = o & 127;
        float acc = bg[c];
        for (int k = 0; k < D; ++k) acc += sx[g * D + k] * wg[(size_t)k * D + c];
        out[o] = (acc > 0.f) ? acc : 0.01f * acc;
    }
}

extern "C" void kernel_launch(void* const* d_in, const int* in_sizes, int n_in,
                              void* d_out, int out_size, void* d_ws, size_t ws_size,
                              hipStream_t stream) {
    const float*     x     = (const float*)d_in[0];
    const long long* ei    = (const long long*)d_in[1];
    const long long* batch = (const long long*)d_in[2];
    const float* w[4] = {(const float*)d_in[4], (const float*)d_in[6],
                         (const float*)d_in[8], (const float*)d_in[10]};
    const float* b[4] = {(const float*)d_in[5], (const float*)d_in[7],
                         (const float*)d_in[9], (const float*)d_in[11]};
    const float* wg = (const float*)d_in[12];
    const float* bg = (const float*)d_in[13];

    int nNodes = in_sizes[0] / D;
    int nEdges = in_sizes[1] / 2;
    float* out = (float*)d_out;

    char* ws = (char*)d_ws;
    size_t nd = (size_t)nNodes * D;
    float*    hA   = (float*)ws;
    float*    hB   = (float*)(ws + nd * 4);
    unsigned* aggu = (unsigned*)(ws + 2 * nd * 4);
    unsigned* xgu  = (unsigned*)(ws + 3 * nd * 4);

    int fillBlocks = (int)((nd + 255) / 256);
    long long sThreads = (long long)nEdges * 32;
    int sBlocks = (int)((sThreads + 255) / 256);
    int gBlocks = (nNodes + 127) / 128;

    const float* hcur = x;
    float* targets[4] = {hA, hB, hA, out};
    for (int s = 0; s < 4; ++s) {
        fill_u32<<<fillBlocks, 256, 0, stream>>>(aggu, NEG_INF_ORD, (int)nd);
        edge_scatter_max<<<sBlocks, 256, 0, stream>>>(hcur, ei, aggu, nEdges);
        gemm_step<<<gBlocks, 256, 0, stream>>>(hcur, aggu, w[s], b[s],
                                               (s > 0) ? 1 : 0, targets[s], nNodes);
        hcur = targets[s];
    }

    fill_u32<<<(NGRAPH * D + 255) / 256, 256, 0, stream>>>(xgu, NEG_INF_ORD, NGRAPH * D);
    pool_max<<<(nNodes + 63) / 64, 128, 0, stream>>>(out, batch, xgu, nNodes);
    xg_final<<<1, 256, 0, stream>>>(xgu, wg, bg, out + nd);
}